// EntropyModel_14525579395262
// MI455X (gfx1250) — compile-verified
//
#include <hip/hip_runtime.h>
#include <math.h>

#define V_ 256
#define D_ 512
#define L_ 4
#define B_ 8
#define S_ 4096
#define ROWS (B_ * S_)          // 32768
#define NCHUNK 32
#define CHUNK (S_ / NCHUNK)     // 128

typedef __attribute__((ext_vector_type(2))) float v2f;
typedef __attribute__((ext_vector_type(8))) float v8f;

__device__ __forceinline__ float softplus_f(float x) {
    // log(1+exp(x)), stable for all x
    return fmaxf(x, 0.0f) + log1pf(expf(-fabsf(x)));
}

// ---------------------------------------------------------------- embedding
__global__ void embed_k(const int* __restrict__ x, const float* __restrict__ emb,
                        float* __restrict__ h) {
    size_t i = (size_t)blockIdx.x * blockDim.x + threadIdx.x;   // over ROWS*D
    int row = (int)(i / D_);
    int d   = (int)(i % D_);
    h[i] = emb[(size_t)x[row] * D_ + d];
}

// ---------------------------------------------------------------- rmsnorm (1 wave / row)
__global__ void rmsnorm_k(const float* __restrict__ h, const float* __restrict__ w,
                          float* __restrict__ out) {
    int wave = (int)((blockIdx.x * blockDim.x + threadIdx.x) >> 5);
    int lane = threadIdx.x & 31;
    const float* hr = h + (size_t)wave * D_;
    float vals[16];
    float ss = 0.0f;
#pragma unroll
    for (int j = 0; j < 16; ++j) {
        float v = hr[lane + j * 32];
        vals[j] = v;
        ss += v * v;
    }
#pragma unroll
    for (int off = 16; off > 0; off >>= 1) ss += __shfl_xor(ss, off, 32);
    float scale = rsqrtf(ss * (1.0f / D_) + 1e-5f);
    float* orow = out + (size_t)wave * D_;
#pragma unroll
    for (int j = 0; j < 16; ++j) {
        int d = lane + j * 32;
        orow[d] = vals[j] * scale * w[d];
    }
}

// ---------------------------------------------------------------- layernorm (1 wave / row)
__global__ void layernorm_k(const float* __restrict__ h, const float* __restrict__ w,
                            const float* __restrict__ b, float* __restrict__ out) {
    int wave = (int)((blockIdx.x * blockDim.x + threadIdx.x) >> 5);
    int lane = threadIdx.x & 31;
    const float* hr = h + (size_t)wave * D_;
    float vals[16];
    float sum = 0.0f;
#pragma unroll
    for (int j = 0; j < 16; ++j) {
        float v = hr[lane + j * 32];
        vals[j] = v;
        sum += v;
    }
#pragma unroll
    for (int off = 16; off > 0; off >>= 1) sum += __shfl_xor(sum, off, 32);
    float mu = sum * (1.0f / D_);
    float ss = 0.0f;
#pragma unroll
    for (int j = 0; j < 16; ++j) {
        float c = vals[j] - mu;
        ss += c * c;
    }
#pragma unroll
    for (int off = 16; off > 0; off >>= 1) ss += __shfl_xor(ss, off, 32);
    float inv = rsqrtf(ss * (1.0f / D_) + 1e-5f);
    float* orow = out + (size_t)wave * D_;
#pragma unroll
    for (int j = 0; j < 16; ++j) {
        int d = lane + j * 32;
        orow[d] = (vals[j] - mu) * inv * w[d] + b[d];
    }
}

// ---------------------------------------------------------------- fp32 WMMA GEMM
// C[M,N] = A[M,K] @ W[N,K]^T (+bias[N]) (+residual[M,N]); wave = 16x64 tile.
// KC is compile-time: the k-loop fully unrolls, every load keeps a constant
// immediate offset (max 2KB < 24-bit field) and the scheduler can hoist loads
// arbitrarily far ahead of their WMMAs (deep clauses, staggered waits).
template <int KC>
__global__ void gemm_k(const float* __restrict__ A, const float* __restrict__ W,
                       const float* __restrict__ bias, const float* __restrict__ residual,
                       float* __restrict__ C, int M, int N) {
    int wave = (int)((blockIdx.x * blockDim.x + threadIdx.x) >> 5);
    int lane = threadIdx.x & 31;
    int nstrips = N >> 6;
    int mtile  = wave / nstrips;
    int nstrip = wave % nstrips;
    if (mtile >= (M >> 4)) return;

    int m16   = lane & 15;
    int mrow  = (mtile << 4) + m16;
    int n0    = nstrip << 6;
    int khalf = (lane >> 4) << 1;       // 0 or 2

    const float* arow = A + (size_t)mrow * KC + khalf;
    const float* w0 = W + (size_t)(n0 + m16)      * KC + khalf;
    const float* w1 = W + (size_t)(n0 + m16 + 16) * KC + khalf;
    const float* w2 = W + (size_t)(n0 + m16 + 32) * KC + khalf;
    const float* w3 = W + (size_t)(n0 + m16 + 48) * KC + khalf;

    v8f acc0 = {}; v8f acc1 = {}; v8f acc2 = {}; v8f acc3 = {};

#pragma unroll
    for (int k = 0; k < KC; k += 4) {
        v2f a  = *(const v2f*)(arow + k);
        v2f b0 = *(const v2f*)(w0 + k);
        v2f b1 = *(const v2f*)(w1 + k);
        v2f b2 = *(const v2f*)(w2 + k);
        v2f b3 = *(const v2f*)(w3 + k);
        acc0 = __builtin_amdgcn_wmma_f32_16x16x4_f32(false, a, false, b0, (short)0, acc0, false, false);
        acc1 = __builtin_amdgcn_wmma_f32_16x16x4_f32(false, a, false, b1, (short)0, acc1, false, false);
        acc2 = __builtin_amdgcn_wmma_f32_16x16x4_f32(false, a, false, b2, (short)0, acc2, false, false);
        acc3 = __builtin_amdgcn_wmma_f32_16x16x4_f32(false, a, false, b3, (short)0, acc3, false, false);
    }

    int mbase = (mtile << 4) + ((lane >> 4) << 3);  // +8 for upper half-lanes
#pragma unroll
    for (int r = 0; r < 8; ++r) {
        size_t rowoff = (size_t)(mbase + r) * N;
        float vs[4] = {acc0[r], acc1[r], acc2[r], acc3[r]};
#pragma unroll
        for (int j = 0; j < 4; ++j) {
            int col = n0 + j * 16 + m16;
            float v = vs[j];
            if (bias)     v += bias[col];
            if (residual) v += residual[rowoff + col];
            C[rowoff + col] = v;
        }
    }
}

// ---------------------------------------------------------------- scan pass 1
// per (chunk,b,d): local cumsum of log_coeffs + local running logcumsumexp.
// Writes a_loc over hg[...,d] and L_loc over hg[...,D+d]; chunk totals to Tsum/Etot.
__global__ void scan1_k(float* __restrict__ hg, float* __restrict__ Tsum,
                        float* __restrict__ Etot) {
    int t = blockIdx.x * blockDim.x + threadIdx.x;  // NCHUNK*B*D threads
    int d  = t % D_;
    int bd = t / D_;
    int b     = bd % B_;
    int chunk = bd / B_;
    size_t base = ((size_t)b * S_ + (size_t)chunk * CHUNK) * (2 * D_);
    float a = 0.0f;
    float mx = -__builtin_inff(), se = 0.0f;
    for (int s = 0; s < CHUNK; ++s) {
        size_t off = base + (size_t)s * (2 * D_);
        float hid  = hg[off + d];
        float gate = hg[off + D_ + d];
        float lc = -softplus_f(gate);                 // log(1-z)
        float lg = (hid >= 0.0f) ? logf(hid + 0.5f) : -softplus_f(-hid);
        float lv = -softplus_f(-gate) + lg;           // log(z*h~)
        a += lc;
        float val = lv - a;
        if (val > mx) { se = se * expf(mx - val) + 1.0f; mx = val; }
        else          { se += expf(val - mx); }
        hg[off + d]      = a;                         // a_loc
        hg[off + D_ + d] = mx + logf(se);             // L_loc
    }
    int cidx = (chunk * B_ + b) * D_ + d;
    Tsum[cidx] = a;
    Etot[cidx] = mx + logf(se);
}

// ---------------------------------------------------------------- scan pass 2
// per (b,d): exclusive prefixes across chunks: Apre (cumsum), Ppre (logsumexp of E_j - Apre_j)
__global__ void scan2_k(const float* __restrict__ Tsum, const float* __restrict__ Etot,
                        float* __restrict__ Apre, float* __restrict__ Ppre) {
    int t = blockIdx.x * blockDim.x + threadIdx.x;   // B*D threads
    int d = t % D_;
    int b = t / D_;
    float A = 0.0f;
    float mx = -__builtin_inff(), se = 0.0f;
    for (int c = 0; c < NCHUNK; ++c) {
        int idx = (c * B_ + b) * D_ + d;
        Apre[idx] = A;
        Ppre[idx] = (se > 0.0f) ? (mx + logf(se)) : -__builtin_inff();
        float Ej = Etot[idx] - A;
        if (Ej > mx) { se = se * expf(mx - Ej) + 1.0f; mx = Ej; }
        else         { se += expf(Ej - mx); }
        A += Tsum[idx];
    }
}

// ---------------------------------------------------------------- scan pass 3
// h_inner = exp( (Apre + a_loc) + logaddexp(Ppre, L_loc - Apre) )
__global__ void scan3_k(const float* __restrict__ hg, const float* __restrict__ Apre,
                        const float* __restrict__ Ppre, float* __restrict__ hin) {
    size_t t = (size_t)blockIdx.x * blockDim.x + threadIdx.x;  // B*S*D
    int d  = (int)(t % D_);
    int rs = (int)(t / D_);
    int s = rs % S_;
    int b = rs / S_;
    int chunk = s / CHUNK;
    size_t off = ((size_t)b * S_ + s) * (2 * D_);
    float aloc = hg[off + d];
    float Lloc = hg[off + D_ + d];
    int cidx = (chunk * B_ + b) * D_ + d;
    float A = Apre[cidx];
    float P = Ppre[cidx];
    float x1 = Lloc - A;
    float m  = fmaxf(P, x1);
    float lse = m + log1pf(expf(fminf(P, x1) - m));   // P=-inf handled: exp(-inf)=0
    hin[t] = expf(A + aloc + lse);
}

// ---------------------------------------------------------------- launch
extern "C" void kernel_launch(void* const* d_in, const int* in_sizes, int n_in,
                              void* d_out, int out_size, void* d_ws, size_t ws_size,
                              hipStream_t stream) {
    const int*   x     = (const int*)d_in[0];
    const float* emb   = (const float*)d_in[1];
    const float* rms_w = (const float*)d_in[2];
    const float* W_hg  = (const float*)d_in[3];
    const float* W_out = (const float*)d_in[4];
    const float* b_out = (const float*)d_in[5];
    const float* ln_w  = (const float*)d_in[6];
    const float* ln_b  = (const float*)d_in[7];
    const float* out_W = (const float*)d_in[8];
    const float* out_b = (const float*)d_in[9];
    float* out = (float*)d_out;

    const size_t nhd = (size_t)ROWS * D_;        // 16.78M floats
    float* ws   = (float*)d_ws;
    float* h    = ws;                            // [B,S,D]
    float* hn   = ws + nhd;                      // [B,S,D] (rmsnorm out / h_inner)
    float* hg   = ws + 2 * nhd;                  // [B,S,2D]
    float* Tsum = ws + 4 * nhd;                  // [NCHUNK,B,D]
    float* Etot = Tsum + (size_t)NCHUNK * B_ * D_;
    float* Apre = Etot + (size_t)NCHUNK * B_ * D_;
    float* Ppre = Apre + (size_t)NCHUNK * B_ * D_;

    const int TB = 256;
    // 1. embedding gather
    embed_k<<<(unsigned)(nhd / TB), TB, 0, stream>>>(x, emb, h);

    for (int l = 0; l < L_; ++l) {
        // 2. rmsnorm (1 wave/row): ROWS waves
        rmsnorm_k<<<ROWS * 32 / TB, TB, 0, stream>>>(h, rms_w + (size_t)l * D_, hn);
        // 3. hg = hn @ W_hg[l]^T   (M=32768, N=1024, K=512): (M/16)*(N/64) waves
        {
            int waves = (ROWS / 16) * (2 * D_ / 64);
            gemm_k<D_><<<waves * 32 / TB, TB, 0, stream>>>(hn, W_hg + (size_t)l * 2 * D_ * D_,
                                                           nullptr, nullptr, hg, ROWS, 2 * D_);
        }
        // 4-6. chunked log-space scan
        scan1_k<<<NCHUNK * B_ * D_ / TB, TB, 0, stream>>>(hg, Tsum, Etot);
        scan2_k<<<B_ * D_ / TB, TB, 0, stream>>>(Tsum, Etot, Apre, Ppre);
        scan3_k<<<(unsigned)(nhd / TB), TB, 0, stream>>>(hg, Apre, Ppre, hn);
        // 7. h = h + hn @ W_out[l]^T + b_out[l]   (M=32768, N=512, K=512)
        {
            int waves = (ROWS / 16) * (D_ / 64);
            gemm_k<D_><<<waves * 32 / TB, TB, 0, stream>>>(hn, W_out + (size_t)l * D_ * D_,
                                                           b_out + (size_t)l * D_, h, h,
                                                           ROWS, D_);
        }
    }

    // 8. layernorm
    layernorm_k<<<ROWS * 32 / TB, TB, 0, stream>>>(h, ln_w, ln_b, hn);
    // 9. logits = hn @ out_W^T + out_b   (M=32768, N=256, K=512)
    {
        int waves = (ROWS / 16) * (V_ / 64);
        gemm_k<D_><<<waves * 32 / TB, TB, 0, stream>>>(hn, out_W, out_b, nullptr, out,
                                                       ROWS, V_);
    }
}